// MixHopLayer_4501125726317
// MI455X (gfx1250) — compile-verified
//
#include <hip/hip_runtime.h>
#include <hip/hip_bf16.h>
#include <stdint.h>

#define C_DIM 128
#define OUT_STRIDE 384      // (HOPS+1)*C
#define LDS_STRIDE 132      // padded row stride for transposed W in LDS
#define WAVES_PER_BLOCK 8

typedef __attribute__((ext_vector_type(2))) float v2f;
typedef __attribute__((ext_vector_type(8))) float v8f;

// ---------------------------------------------------------------------------
// Hardware f32 atomic add (non-returning -> STOREcnt path, no CAS loop)
// ---------------------------------------------------------------------------
__device__ __forceinline__ void atomic_add_f32_hw(float* p, float v) {
  asm volatile("global_atomic_add_f32 %0, %1, off"
               :
               : "v"((uint64_t)(uintptr_t)p), "v"(v)
               : "memory");
}

// ---------------------------------------------------------------------------
// Zero-fill (atomic-accumulate destinations must start at 0; harness poisons)
// ---------------------------------------------------------------------------
__global__ __launch_bounds__(256) void mixhop_zero_kernel(float4* __restrict__ p,
                                                          long n4) {
  long i = (long)blockIdx.x * blockDim.x + threadIdx.x;
  const long stride = (long)gridDim.x * blockDim.x;
  const float4 z = make_float4(0.f, 0.f, 0.f, 0.f);
  for (; i < n4; i += stride) p[i] = z;
}

// ---------------------------------------------------------------------------
// Dense GEMM via V_WMMA_F32_16X16X4_F32:  H_j = x @ W[j] + b[j]
//
// Block = 256 threads (8 waves); each wave computes a 16x128 output tile.
// W[j] is staged TRANSPOSED into LDS once per block (row stride 132 floats:
// 8B-aligned b64 fragment loads, all 64 LDS banks hit exactly once per wave).
//
// Fragment layouts (CDNA5 ISA 7.12.2, 32-bit data):
//   A 16x4 : lane L holds row m=L%16; VGPR v holds K = v + 2*(L/16)
//   B 4x16 : lane L holds col n=L%16; VGPR v holds K = v + 2*(L/16)
//   C 16x16: lane L holds col n=L%16; VGPR v holds row M = v + 8*(L/16)
// ---------------------------------------------------------------------------
__global__ __launch_bounds__(256) void mixhop_gemm_kernel(
    const float* __restrict__ x,    // [N, C]
    const float* __restrict__ W,    // [3, C, C]
    const float* __restrict__ bias, // [3, C]
    float* __restrict__ out,        // d_out base (j==0 dest, stride OUT_STRIDE)
    float* __restrict__ h1,         // ws [N, C]
    float* __restrict__ h2,         // ws [N, C]
    int n, int ntiles) {
  __shared__ float ldsWT[C_DIM * LDS_STRIDE];  // W[j]^T, padded: 66 KB

  const int tid   = threadIdx.x;
  const int j     = blockIdx.y;
  const float* Wj = W + (size_t)j * C_DIM * C_DIM;

  // ---- stage W[j]^T into LDS: read row-major float4 (coalesced), scatter ----
  {
    // 128x128 floats = 4096 float4 chunks; 16 chunks per thread
#pragma unroll
    for (int i = 0; i < 16; ++i) {
      const int c  = tid + i * 256;       // chunk id
      const int k  = c >> 5;              // W row
      const int n0 = (c & 31) * 4;        // W col base
      const float4 w4 = *(const float4*)(Wj + (size_t)k * C_DIM + n0);
      ldsWT[(n0 + 0) * LDS_STRIDE + k] = w4.x;
      ldsWT[(n0 + 1) * LDS_STRIDE + k] = w4.y;
      ldsWT[(n0 + 2) * LDS_STRIDE + k] = w4.z;
      ldsWT[(n0 + 3) * LDS_STRIDE + k] = w4.w;
    }
  }
  __syncthreads();

  const int wave  = tid >> 5;
  const int lane  = tid & 31;
  const int tile  = blockIdx.x * WAVES_PER_BLOCK + wave;
  if (tile >= ntiles) return;              // whole wave exits: EXEC full for WMMA

  const int m     = lane & 15;
  const int khalf = lane >> 4;             // 0 or 1
  const int row0  = tile * 16;

  int r = row0 + m;
  if (r >= n) r = n - 1;                   // clamp reads; stores predicated below
  const float* arow = x + (size_t)r * C_DIM + 2 * khalf;

  v8f acc[8];
#pragma unroll
  for (int t = 0; t < 8; ++t) {
    v8f z = {0.f, 0.f, 0.f, 0.f, 0.f, 0.f, 0.f, 0.f};
    acc[t] = z;
  }

  // base LDS pointer for this lane's B fragments (n = t*16 + m, K = k + 2*khalf)
  const float* bt0 = &ldsWT[(size_t)m * LDS_STRIDE + 2 * khalf];

  for (int k = 0; k < C_DIM; k += 4) {
    v2f a;
    a.x = arow[k + 0];                     // A[m][k + 2*khalf + 0]
    a.y = arow[k + 1];                     // A[m][k + 2*khalf + 1]
    const float* bt = bt0 + k;
#pragma unroll
    for (int t = 0; t < 8; ++t) {
      const v2f bb = *(const v2f*)(bt + t * 16 * LDS_STRIDE);  // ds_load_b64
      acc[t] = __builtin_amdgcn_wmma_f32_16x16x4_f32(
          false, a, false, bb, (short)0, acc[t], false, false);
    }
  }

  float* dst;
  int dstStride;
  if (j == 0) { dst = out; dstStride = OUT_STRIDE; }
  else if (j == 1) { dst = h1; dstStride = C_DIM; }
  else { dst = h2; dstStride = C_DIM; }

#pragma unroll
  for (int t = 0; t < 8; ++t) {
    const int col = t * 16 + m;
    const float bv = bias[j * C_DIM + col];
#pragma unroll
    for (int vr = 0; vr < 8; ++vr) {
      const int row = row0 + vr + 8 * khalf;
      if (row < n) dst[(size_t)row * dstStride + col] = acc[t][vr] + bv;
    }
  }
}

// ---------------------------------------------------------------------------
// SpMM scatter:  dst[row[e], :] += val[e] * h[col[e], :]
// One wave per edge: 32 lanes x float4 = 128 channels.  h rows are L2-resident
// (25.6 MB << 192 MB L2).  Hardware global_atomic_add_f32, non-returning.
// ---------------------------------------------------------------------------
__global__ __launch_bounds__(256) void mixhop_spmm_kernel(
    const int* __restrict__ erow, const int* __restrict__ ecol,
    const float* __restrict__ eval,
    const float* __restrict__ h,     // [N, C]
    float* __restrict__ dst,         // accumulate dest (col offset pre-folded)
    int dstStride, int E) {
  const int e = blockIdx.x * 8 + (threadIdx.x >> 5);
  if (e >= E) return;
  const int lane = threadIdx.x & 31;
  const int c = ecol[e];
  const int r = erow[e];
  const float v = eval[e];
  const float4 s = *(const float4*)(h + (size_t)c * C_DIM + lane * 4);
  float* d = dst + (size_t)r * dstStride + lane * 4;
  atomic_add_f32_hw(d + 0, v * s.x);
  atomic_add_f32_hw(d + 1, v * s.y);
  atomic_add_f32_hw(d + 2, v * s.z);
  atomic_add_f32_hw(d + 3, v * s.w);
}

// ---------------------------------------------------------------------------
// Launch
// ---------------------------------------------------------------------------
extern "C" void kernel_launch(void* const* d_in, const int* in_sizes, int n_in,
                              void* d_out, int out_size, void* d_ws, size_t ws_size,
                              hipStream_t stream) {
  const float* x    = (const float*)d_in[0];   // [N, 128]
  const float* W    = (const float*)d_in[1];   // [3, 128, 128]
  const float* b    = (const float*)d_in[2];   // [3, 128]
  const float* eval = (const float*)d_in[3];   // [E]
  const int*   erow = (const int*)d_in[4];     // [E]
  const int*   ecol = (const int*)d_in[5];     // [E]
  float* out = (float*)d_out;                  // [N, 384]

  const int N = in_sizes[0] / C_DIM;           // 50000
  const int E = in_sizes[3];                   // 600000

  // Workspace: H1 | H2 | T2, each [N, 128] fp32
  float* h1 = (float*)d_ws;
  float* h2 = h1 + (size_t)N * C_DIM;
  float* t2 = h2 + (size_t)N * C_DIM;

  // 1) zero atomic-accumulation destinations (d_out fully, and T2)
  {
    const long out4 = (long)N * OUT_STRIDE / 4;
    mixhop_zero_kernel<<<2048, 256, 0, stream>>>((float4*)out, out4);
    const long t24 = (long)N * C_DIM / 4;
    mixhop_zero_kernel<<<1024, 256, 0, stream>>>((float4*)t2, t24);
  }

  // 2) three GEMMs: j=0 -> out[:, 0:128]; j=1 -> H1; j=2 -> H2
  {
    const int ntiles = (N + 15) / 16;                          // 3125
    dim3 grid((ntiles + WAVES_PER_BLOCK - 1) / WAVES_PER_BLOCK, 3);  // 391 x 3
    mixhop_gemm_kernel<<<grid, 256, 0, stream>>>(x, W, b, out, h1, h2, N, ntiles);
  }

  // 3) SpMM chain (stream-ordered):
  //    H1 -> out[:,128:256];  H2 -> T2;  T2 -> out[:,256:384]
  {
    const int blocks = (E + 7) / 8;            // 8 edges (waves) per 256-thread block
    mixhop_spmm_kernel<<<blocks, 256, 0, stream>>>(erow, ecol, eval, h1,
                                                   out + 128, OUT_STRIDE, E);
    mixhop_spmm_kernel<<<blocks, 256, 0, stream>>>(erow, ecol, eval, h2,
                                                   t2, C_DIM, E);
    mixhop_spmm_kernel<<<blocks, 256, 0, stream>>>(erow, ecol, eval, t2,
                                                   out + 256, OUT_STRIDE, E);
  }
}